// RFInterDAS_67585605370339
// MI455X (gfx1250) — compile-verified
//
#include <hip/hip_runtime.h>
#include <stdint.h>

// Problem dimensions (match reference)
#define N_ANG   8
#define N_EL    128
#define N_SAMP  2048
#define NPIX    (384 * 192)          // 73728 pixels
#define THREADS 256                  // 8 wave32 waves per block
#define PPT     4                    // pixels per thread
#define PPB     (THREADS * PPT)      // 1024 pixels per block -> 72 tiles
#define APB     2                    // angles per block -> 4 angle groups
#define TRACE_BYTES  (N_SAMP * 4)        // 8192 B per (angle,element) trace
#define STAGE_FLOATS (APB * N_SAMP)      // 4096 floats staged per element
#define STAGE_BYTES  (STAGE_FLOATS * 4)  // 16384 B
#define CHUNKS       (STAGE_BYTES / 16)  // 1024 x 16B chunks
#define LOADS_PER_THREAD (CHUNKS / THREADS) // 4 async b128 per thread/stage

#define CLAMP_HI 2046.999f           // float(N_SAMP - 1.001), matches reference

__global__ __launch_bounds__(THREADS)
void das_async_lds_kernel(const float* __restrict__ rf,
                          const float* __restrict__ t0,
                          const float* __restrict__ d_tx,
                          const float* __restrict__ d_rx,
                          const float* __restrict__ fs_p,
                          const float* __restrict__ c0_p,
                          const float* __restrict__ apod,
                          float* __restrict__ out)
{
    extern __shared__ float sm[];   // 2 * STAGE_FLOATS floats (32 KB), double buffer

    const int tid  = threadIdx.x;
    const int tile = blockIdx.x;          // 0..71  pixel tile
    const int ag   = blockIdx.y;          // 0..3   angle group
    const int a0   = ag * APB;

    const float fs    = fs_p[0];
    const float scale = fs / c0_p[0];     // samples per meter of path

    int p[PPT];
#pragma unroll
    for (int k = 0; k < PPT; ++k)
        p[k] = tile * PPB + k * THREADS + tid;

    // Pre-fold: s_idx = (t0*fs + d_tx*scale) + d_rx*scale
    float dtxs[APB][PPT];
#pragma unroll
    for (int al = 0; al < APB; ++al) {
        const float base = t0[a0 + al] * fs;          // uniform -> scalar load
#pragma unroll
        for (int k = 0; k < PPT; ++k)
            dtxs[al][k] = fmaf(d_tx[(a0 + al) * NPIX + p[k]], scale, base);
    }

    float acc[APB][PPT];
#pragma unroll
    for (int al = 0; al < APB; ++al)
#pragma unroll
        for (int k = 0; k < PPT; ++k) acc[al][k] = 0.0f;

    const uint32_t  lds_base = (uint32_t)(uintptr_t)sm;  // LDS offset (flat low 32 bits)
    const uintptr_t rf_addr  = (uintptr_t)rf;

    // Stage the APB traces of element e into LDS buffer b via the async engine.
    auto stage = [&](int e, int b) {
#pragma unroll
        for (int i = 0; i < LOADS_PER_THREAD; ++i) {
            const int c = tid + i * THREADS;              // 16B chunk id, 0..1023
            const int al = c >> 9;                        // which of the 2 traces
            const uint32_t goff = (uint32_t)(((a0 + al) * N_EL + e) * TRACE_BYTES)
                                + (uint32_t)((c & 511) * 16);
            const uint32_t loff = lds_base + (uint32_t)(b * STAGE_BYTES + c * 16);
            asm volatile("global_load_async_to_lds_b128 %0, %1, %2"
                         :: "v"(loff), "v"(goff), "s"(rf_addr)
                         : "memory");
        }
    };

    stage(0, 0);  // prologue

    for (int e = 0; e < N_EL; ++e) {
        // Prefetch next element into the other buffer, then wait only for
        // the CURRENT buffer's 4 in-order async loads (slack wait keeps the
        // prefetch in flight).
        if (e + 1 < N_EL) {
            stage(e + 1, (e + 1) & 1);
            asm volatile("s_wait_asynccnt 4" ::: "memory");
        } else {
            asm volatile("s_wait_asynccnt 0" ::: "memory");
        }
        __syncthreads();   // all waves' staged portions visible

        const float* buf = sm + (e & 1) * STAGE_FLOATS;

        float rx[PPT], w[PPT];
#pragma unroll
        for (int k = 0; k < PPT; ++k) {
            rx[k] = d_rx[e * NPIX + p[k]];
            w[k]  = apod[e * NPIX + p[k]];
        }

#pragma unroll
        for (int al = 0; al < APB; ++al) {
            const float* tr = buf + al * N_SAMP;
#pragma unroll
            for (int k = 0; k < PPT; ++k) {
                float s  = fmaf(rx[k], scale, dtxs[al][k]);          // 1 FMA
                s        = __builtin_amdgcn_fmed3f(s, 0.0f, CLAMP_HI); // v_med3_f32
                int   il = (int)s;                                   // trunc == floor (s>=0)
                float fr = __builtin_amdgcn_fractf(s);               // v_fract_f32
                float lo = tr[il];        // LDS gather
                float hi = tr[il + 1];    // LDS gather
                acc[al][k] = fmaf(w[k], fmaf(fr, hi - lo, lo), acc[al][k]);
            }
        }
        __syncthreads();   // buffer is overwritten by the next iteration's prefetch
    }

#pragma unroll
    for (int al = 0; al < APB; ++al)
#pragma unroll
        for (int k = 0; k < PPT; ++k)
            out[(a0 + al) * NPIX + p[k]] = acc[al][k];
}

extern "C" void kernel_launch(void* const* d_in, const int* in_sizes, int n_in,
                              void* d_out, int out_size, void* d_ws, size_t ws_size,
                              hipStream_t stream) {
    (void)in_sizes; (void)n_in; (void)out_size; (void)d_ws; (void)ws_size;
    const float* rf   = (const float*)d_in[0];
    const float* t0   = (const float*)d_in[1];
    const float* d_tx = (const float*)d_in[2];
    const float* d_rx = (const float*)d_in[3];
    const float* fs   = (const float*)d_in[4];
    const float* c0   = (const float*)d_in[5];
    const float* apod = (const float*)d_in[6];
    float* out = (float*)d_out;

    dim3 grid(NPIX / PPB, N_ANG / APB);   // (72, 4) = 288 blocks, 2304 waves
    das_async_lds_kernel<<<grid, THREADS, 2 * STAGE_BYTES, stream>>>(
        rf, t0, d_tx, d_rx, fs, c0, apod, out);
}